// UniversalMVGLayer_41807211659417
// MI455X (gfx1250) — compile-verified
//
#include <hip/hip_runtime.h>
#include <hip/hip_bf16.h>
#include <math.h>

typedef float v2f __attribute__((ext_vector_type(2)));
typedef float v8f __attribute__((ext_vector_type(8)));

#define H_      121
#define W_      121
#define HW      14641      // 121*121
#define B_      8
#define E_      225
#define EPAD    240        // pad electrodes to multiple of 16 per batch
#define NSLOT   (B_ * EPAD)        // 1920 (b,e) slots
#define NTILE_E (EPAD / 16)        // 15 electrode tiles per batch
#define NTILE_P ((HW + 15) / 16)   // 916 pixel tiles
#define WAVES_PER_BLOCK 8

// ---------------------------------------------------------------------------
// Phase 1: per-(batch, electrode) coefficient setup.
// Slot = 8 floats [g0..g5, 0, 0] where
//   t3_eff(px,py) = g0*px^2 + g1*px*py + g2*py^2 + g3*px + g4*py + g5
// and bright is folded into g5:  g5 -= 2*ln(bright)  so that
//   exp(-0.5*t3_eff) == exp(-0.5*t3) * bright.
// Padded slots (e >= 225) and bright==0 slots contribute ~0 (< 1e-30).
// The trailing [0,0] pair doubles as the K=6,7 zero-pad for the upper
// lane-half's second WMMA B-fragment -> all B loads are unconditional b64s.
// ---------------------------------------------------------------------------
__global__ __launch_bounds__(256) void mvg_setup_kernel(
    const float* __restrict__ stim,    // (8,225,3)
    const float* __restrict__ params,  // (8,13)
    const float* __restrict__ elec_x,  // (225)
    const float* __restrict__ elec_y,  // (225)
    const float* __restrict__ slopes,  // (200,200)
    float* __restrict__ G)             // (1920,8) workspace
{
    int idx = blockIdx.x * blockDim.x + threadIdx.x;
    if (idx >= NSLOT) return;
    int b = idx / EPAD;
    int e = idx % EPAD;

    float g0 = 0.f, g1 = 0.f, g2 = 0.f, g3 = 0.f, g4 = 0.f;
    float g5 = 138.0f;   // pad slots: exp(-69) ~ 1e-30 -> 0 contribution
    if (e < E_) {
        const float* pp = params + b * 13;
        float rho = pp[0], lam = pp[1], osc = pp[2];
        float a0 = pp[3], a1 = pp[4], a2 = pp[5], a3 = pp[6], a4 = pp[7];
        float impx = pp[8], impy = pp[9], rot = pp[10], locx = pp[11];

        const float* st = stim + (b * E_ + e) * 3;
        float freq = st[0], amp = st[1], pdur = st[2];

        float cr = cosf(rot), sr = sinf(rot);
        float exv = elec_x[e] * cr - elec_y[e] * sr + impx;
        float eyv = elec_x[e] * sr + elec_y[e] * cr + impy;

        // bilinear interp of slopes at (qx, qy)
        float offx = locx - 4340.0f;             // OD_OFF_X = 15.5*280
        const float INV_STEP = 199.0f / 8400.0f; // 1/STEP_X == 1/STEP_Y
        float qx = (exv - offx + 4200.0f) * INV_STEP;  // XLOW = -4200
        float qy = (eyv + 4200.0f) * INV_STEP;         // YLOW = -4200
        float fx = fminf(fmaxf(floorf(qx), 0.0f), 198.0f);
        float fy = fminf(fmaxf(floorf(qy), 0.0f), 198.0f);
        int ixi = (int)fx, iyi = (int)fy;
        float ax = fminf(fmaxf(qx - fx, 0.0f), 1.0f);
        float ay = fminf(fmaxf(qy - fy, 0.0f), 1.0f);
        const float* s0 = slopes + iyi * 200 + ixi;
        float v00 = s0[0], v01 = s0[1], v10 = s0[200], v11 = s0[201];
        float top = v00 + ax * (v01 - v00);
        float bot = v10 + ax * (v11 - v10);
        float th  = top + ay * (bot - top);

        const float PI_F = 3.14159265358979323846f;
        th = (th < -0.5f * PI_F) ? th + PI_F : th;
        th *= osc;

        float rho_s = fmaxf(rho * amp * a3, 1.0f);
        float lam_s = lam * powf(0.45f, -a4) * powf(pdur, a4);
        lam_s = fminf(fmaxf(lam_s, 0.0f), 0.99f);
        float br = (amp > 0.25f) ? (a0 * powf(fmaxf(amp, 1e-5f), a1) + a2 * freq)
                                 : 0.0f;

        float t2 = sqrtf(1.0f - lam_s * lam_s);
        const float T1 = 12.566370614359172f;   // 4*pi
        float sy_ = rho_s / (T1 * t2);
        float sx_ = rho_s * t2 / T1;
        float sn = sinf(th), cs = cosf(th);
        float cov00 = sx_ * cs * cs + sy_ * sn * sn;
        float cov01 = (sx_ - sy_) * sn * cs;
        float cov11 = sx_ * sn * sn + sy_ * cs * cs;
        float det = cov00 * cov11 - cov01 * cov01;
        float inv00 = cov11 / det, inv01 = -cov01 / det, inv11 = cov00 / det;

        float cx = exv * (1.0f / 70.0f) + 60.0f;   // (ex/280+15)/0.25
        float cy = 61.0f - eyv * (1.0f / 70.0f);   // 121-(ey/280+15)/0.25

        g0 = inv00;
        g1 = 2.0f * inv01;
        g2 = inv11;
        g3 = -2.0f * (cx * inv00 + cy * inv01);
        g4 = -2.0f * (cy * inv11 + cx * inv01);
        g5 = cx * cx * inv00 + 2.0f * cx * cy * inv01 + cy * cy * inv11
             - 2.0f * logf(fmaxf(br, 1e-30f));     // bright folded in
    }
    float4* o = (float4*)(G + idx * 8);
    o[0] = make_float4(g0, g1, g2, g3);
    o[1] = make_float4(g4, g5, 0.0f, 0.0f);
}

// ---------------------------------------------------------------------------
// Phase 2: T3 = F(P x 6) * G^T(6 x BE) via V_WMMA_F32_16X16X4_F32 (K padded
// to 8 = two chained WMMAs), then sum exp(-0.5*T3) over electrode columns.
// One wave per 16-pixel tile; coefficient table staged in LDS per block.
// ---------------------------------------------------------------------------
__global__ __launch_bounds__(256) void mvg_accum_kernel(
    const float* __restrict__ G, float* __restrict__ out)
{
    __shared__ float4 gs4[NSLOT * 2];   // 1920 slots * 32B = 60 KB in LDS
    const float* gs = (const float*)gs4;

    // cooperative stage of the whole coefficient table into LDS
    const float4* src = (const float4*)G;
    for (int i = threadIdx.x; i < NSLOT * 2; i += blockDim.x) gs4[i] = src[i];
    __syncthreads();

    int wave = threadIdx.x >> 5;
    int lane = threadIdx.x & 31;
    int tile = blockIdx.x * WAVES_PER_BLOCK + wave;
    if (tile >= NTILE_P) return;   // whole wave exits together (EXEC all-1s for WMMA)

    int half = lane >> 4;   // selects K-pair (A) / K-rows (B) per f32 WMMA layout
    int m    = lane & 15;   // A: row M within tile; B: column N within tile

    // pixel features for row M = m of this tile
    int p  = tile * 16 + m;
    int pc = p < HW ? p : (HW - 1);          // clamp; writes guarded below
    int x  = pc % W_;
    int y  = pc / W_;
    float px = (float)x;
    float py = (float)(120 - y);             // flip(axis=1) folded in
    // f = [px^2, px*py, py^2, px, py, 1, 0, 0]
    v2f a1, a2;
    if (half == 0) { a1.x = px * px; a1.y = px * py; a2.x = py;  a2.y = 1.0f; }
    else           { a1.x = py * py; a1.y = px;      a2.x = 0.f; a2.y = 0.f; }

    for (int b = 0; b < B_; ++b) {
        // lane's B-fragment base: slot (b*EPAD + m), float offset half*2
        const float* gp = gs + ((b * EPAD + m) * 8) + half * 2;
        v8f acc = {0.f, 0.f, 0.f, 0.f, 0.f, 0.f, 0.f, 0.f};
#pragma unroll
        for (int t = 0; t < NTILE_E; ++t) {
            v2f b1 = *(const v2f*)(gp + t * 128);      // K=2h,2h+1 : (g0,g1)/(g2,g3)
            v2f b2 = *(const v2f*)(gp + t * 128 + 4);  // K=4+2h    : (g4,g5)/(0,0)

            v8f c = {0.f, 0.f, 0.f, 0.f, 0.f, 0.f, 0.f, 0.f};
            c = __builtin_amdgcn_wmma_f32_16x16x4_f32(
                    false, a1, false, b1, (short)0, c, false, false);
            c = __builtin_amdgcn_wmma_f32_16x16x4_f32(
                    false, a2, false, b2, (short)0, c, false, false);
#pragma unroll
            for (int r = 0; r < 8; ++r)
                acc[r] += __expf(-0.5f * c[r]);
        }
        // sum the 16 electrode columns: butterfly within each 16-lane half
#pragma unroll
        for (int r = 0; r < 8; ++r) {
            float v = acc[r];
            v += __shfl_xor(v, 1, 32);
            v += __shfl_xor(v, 2, 32);
            v += __shfl_xor(v, 4, 32);
            v += __shfl_xor(v, 8, 32);
            acc[r] = v;
        }
        // D layout: VGPR r -> row r (lanes 0-15) / row 8+r (lanes 16-31)
        if (m == 0) {
            int rowBase = tile * 16 + half * 8;
#pragma unroll
            for (int r = 0; r < 8; ++r) {
                int pp = rowBase + r;
                if (pp < HW) out[b * HW + pp] = acc[r];
            }
        }
    }
}

extern "C" void kernel_launch(void* const* d_in, const int* in_sizes, int n_in,
                              void* d_out, int out_size, void* d_ws, size_t ws_size,
                              hipStream_t stream) {
    const float* stim    = (const float*)d_in[0];  // (8,225,3)
    const float* params  = (const float*)d_in[1];  // (8,13)
    const float* elec_x  = (const float*)d_in[2];  // (1,225)
    const float* elec_y  = (const float*)d_in[3];  // (1,225)
    const float* slopes  = (const float*)d_in[4];  // (200,200)
    // d_in[5] pixelgrid: indices are recomputed analytically on device
    (void)in_sizes; (void)n_in; (void)out_size; (void)ws_size;

    float* G   = (float*)d_ws;      // 1920*8 floats = 61440 B
    float* out = (float*)d_out;     // 8*121*121 floats

    mvg_setup_kernel<<<(NSLOT + 255) / 256, 256, 0, stream>>>(
        stim, params, elec_x, elec_y, slopes, G);

    int nblocks = (NTILE_P + WAVES_PER_BLOCK - 1) / WAVES_PER_BLOCK;  // 115
    mvg_accum_kernel<<<nblocks, 256, 0, stream>>>(G, out);
}